// PSA_43404939493450
// MI455X (gfx1250) — compile-verified
//
#include <hip/hip_runtime.h>
#include <math.h>

#define B_  16
#define C_  64
#define HW_ 65536

typedef __attribute__((ext_vector_type(2))) float v2f;
typedef __attribute__((ext_vector_type(8))) float v8f;

// ---------------------------------------------------------------------------
// K1: colsum_x[b,c] = sum over HW of x[b,c,:]   (one block per (b,c) row)
// ---------------------------------------------------------------------------
__global__ void __launch_bounds__(256)
psa_colsum(const float* __restrict__ x, float* __restrict__ colsum) {
  const int row = blockIdx.x;  // b*C + c
  const float4* p = reinterpret_cast<const float4*>(x + (size_t)row * HW_);
  float s = 0.f;
  for (int i = threadIdx.x; i < HW_ / 4; i += 256) {
    float4 v = p[i];
    s += (v.x + v.y) + (v.z + v.w);
  }
  __shared__ float red[256];
  red[threadIdx.x] = s;
  __syncthreads();
  for (int off = 128; off > 0; off >>= 1) {
    if ((int)threadIdx.x < off) red[threadIdx.x] += red[threadIdx.x + off];
    __syncthreads();
  }
  if (threadIdx.x == 0) colsum[row] = red[0];
}

// ---------------------------------------------------------------------------
// K2: colsum_y = W @ colsum_x + bias*HW ;  w2[b,c] = sum_o W[o,c]*colsum_y[b,o]
//     bias2[b] = sum_o bias[o]*colsum_y[b,o]       (single 1024-thread block)
// ---------------------------------------------------------------------------
__global__ void __launch_bounds__(1024)
psa_derived(const float* __restrict__ pw_w, const float* __restrict__ pw_b,
            const float* __restrict__ colsum_x, float* __restrict__ colsum_y,
            float* __restrict__ w2, float* __restrict__ bias2) {
  __shared__ float s_csy[B_ * C_];
  const int t = threadIdx.x;        // 0..1023
  const int b = t >> 6, o = t & 63;
  float acc = 0.f;
  for (int c = 0; c < C_; ++c) acc += pw_w[o * C_ + c] * colsum_x[b * C_ + c];
  acc += pw_b[o] * (float)HW_;
  colsum_y[t] = acc;
  s_csy[t] = acc;
  __syncthreads();
  const int c = o;
  float acc2 = 0.f;
  for (int o2 = 0; o2 < C_; ++o2) acc2 += pw_w[o2 * C_ + c] * s_csy[b * C_ + o2];
  w2[t] = acc2;
  if (o == 0) {
    float acc3 = 0.f;
    for (int o2 = 0; o2 < C_; ++o2) acc3 += pw_b[o2] * s_csy[b * C_ + o2];
    bias2[b] = acc3;
  }
}

// ---------------------------------------------------------------------------
// K3: score_x/score_y per pixel (4 pixels/thread, float4) + sum-of-squares
// ---------------------------------------------------------------------------
__global__ void __launch_bounds__(256)
psa_scores(const float* __restrict__ x,
           const float* __restrict__ colsum_x, const float* __restrict__ w2,
           const float* __restrict__ bias2,
           float* __restrict__ score_x, float* __restrict__ score_y,
           float* __restrict__ sumsq_x, float* __restrict__ sumsq_y) {
  const int b = blockIdx.y;
  const int vecIdx = blockIdx.x * 256 + threadIdx.x;  // float4 index in [0, HW/4)
  __shared__ float s_cs[C_], s_w2[C_];
  if (threadIdx.x < C_) {
    s_cs[threadIdx.x] = colsum_x[b * C_ + threadIdx.x];
    s_w2[threadIdx.x] = w2[b * C_ + threadIdx.x];
  }
  __syncthreads();
  const float* xb = x + (size_t)b * C_ * HW_;
  float4 sx = {0.f, 0.f, 0.f, 0.f}, sy = {0.f, 0.f, 0.f, 0.f};
  for (int c = 0; c < C_; ++c) {
    float4 v = reinterpret_cast<const float4*>(xb + (size_t)c * HW_)[vecIdx];
    const float wc = s_cs[c], w2c = s_w2[c];
    sx.x += v.x * wc;  sx.y += v.y * wc;  sx.z += v.z * wc;  sx.w += v.w * wc;
    sy.x += v.x * w2c; sy.y += v.y * w2c; sy.z += v.z * w2c; sy.w += v.w * w2c;
  }
  const float bb = bias2[b];
  sy.x += bb; sy.y += bb; sy.z += bb; sy.w += bb;
  reinterpret_cast<float4*>(score_x + (size_t)b * HW_)[vecIdx] = sx;
  reinterpret_cast<float4*>(score_y + (size_t)b * HW_)[vecIdx] = sy;

  float lx = sx.x * sx.x + sx.y * sx.y + sx.z * sx.z + sx.w * sx.w;
  float ly = sy.x * sy.x + sy.y * sy.y + sy.z * sy.z + sy.w * sy.w;
  __shared__ float rx[256], ry[256];
  rx[threadIdx.x] = lx;
  ry[threadIdx.x] = ly;
  __syncthreads();
  for (int off = 128; off > 0; off >>= 1) {
    if ((int)threadIdx.x < off) {
      rx[threadIdx.x] += rx[threadIdx.x + off];
      ry[threadIdx.x] += ry[threadIdx.x + off];
    }
    __syncthreads();
  }
  if (threadIdx.x == 0) {
    atomicAdd(&sumsq_x[b], rx[0]);
    atomicAdd(&sumsq_y[b], ry[0]);
  }
}

// ---------------------------------------------------------------------------
// K4: inverse norms (EPS-clamped, matches F.normalize semantics)
// ---------------------------------------------------------------------------
__global__ void psa_norms(const float* __restrict__ ssx, const float* __restrict__ ssy,
                          float* __restrict__ inx, float* __restrict__ iny) {
  const int b = threadIdx.x;
  if (b < B_) {
    inx[b] = 1.f / fmaxf(sqrtf(ssx[b]), 1e-12f);
    iny[b] = 1.f / fmaxf(sqrtf(ssy[b]), 1e-12f);
  }
}

// ---------------------------------------------------------------------------
// K5: fused 1x1-conv (WMMA f32 16x16x4) + attention gate + output
//   grid = (HW/128, B), block = 256 (8 waves): wave = mtile(4) x pixelgroup(2)
// ---------------------------------------------------------------------------
__global__ void __launch_bounds__(256)
psa_fused(const float* __restrict__ x,
          const float* __restrict__ pw_w, const float* __restrict__ pw_b,
          const float* __restrict__ score_x, const float* __restrict__ score_y,
          const float* __restrict__ inv_nx, const float* __restrict__ inv_ny,
          float* __restrict__ out) {
  const int b = blockIdx.y;
  const int pix0 = blockIdx.x * 128;

  __shared__ float sW[C_ * C_];     // 16 KB: full 64x64 weight, row-major [o][c]
  __shared__ float sBias[C_];
  __shared__ float sAtt[128];

  for (int i = threadIdx.x; i < C_ * C_; i += 256) sW[i] = pw_w[i];
  if (threadIdx.x < C_) sBias[threadIdx.x] = pw_b[threadIdx.x];

  // Attention gate for this block's 128 pixels: softmax over batch dim, sigmoid
  if (threadIdx.x < 128) {
    const int pix = pix0 + threadIdx.x;
    float d[B_];
    float my_sy = 0.f;
    float m = -INFINITY;
    for (int bb = 0; bb < B_; ++bb) {
      const float sv  = score_x[(size_t)bb * HW_ + pix] * inv_nx[bb];
      const float syv = score_y[(size_t)bb * HW_ + pix] * inv_ny[bb];
      if (bb == b) my_sy = syv;
      const float dd = syv - sv;
      d[bb] = dd;
      m = fmaxf(m, dd);
    }
    float sum = 0.f;
    for (int bb = 0; bb < B_; ++bb) sum += __expf(d[bb] - m);
    const float raise = __expf(d[b] - m) / sum;
    const float a = my_sy * (1.f + raise);
    sAtt[threadIdx.x] = 1.f / (1.f + __expf(-a));
  }
  __syncthreads();

  const int wave  = threadIdx.x >> 5;
  const int lane  = threadIdx.x & 31;
  const int mtile = wave & 3;          // which 16 output channels
  const int pgrp  = wave >> 2;         // which 64-pixel group
  const int mrow  = lane & 15;         // A-matrix M row (ISA 16x4 layout)
  const int khalf = lane >> 4;         // lanes 16-31 hold K=2,3
  const int pixl  = pix0 + pgrp * 64 + (lane & 15);  // B/D pixel for this lane

  const float* xb = x + (size_t)b * C_ * HW_;
  const float* wrow = &sW[(mtile * 16 + mrow) * C_];

  v8f acc[4] = {};  // 4 ntiles of 16 pixels each

  #pragma unroll 4
  for (int k0 = 0; k0 < C_; k0 += 4) {
    const int cb = k0 + 2 * khalf;
    v2f a;                              // A 16x4 fragment (weights)
    a.x = wrow[cb];
    a.y = wrow[cb + 1];
    const float* xc0 = xb + (size_t)cb * HW_;
    #pragma unroll
    for (int j = 0; j < 4; ++j) {
      v2f bf;                           // B 4x16 fragment (activations)
      bf.x = xc0[j * 16 + pixl];
      bf.y = xc0[HW_ + j * 16 + pixl];
      acc[j] = __builtin_amdgcn_wmma_f32_16x16x4_f32(
          false, a, false, bf, (short)0, acc[j], false, false);
    }
  }

  // Epilogue: bias + gate + store (C/D layout: VGPR r -> M = r + 8*khalf)
  float* ob = out + (size_t)b * C_ * HW_;
  #pragma unroll
  for (int j = 0; j < 4; ++j) {
    const float att = sAtt[pgrp * 64 + j * 16 + (lane & 15)];
    #pragma unroll
    for (int r = 0; r < 8; ++r) {
      const int o = mtile * 16 + r + 8 * khalf;
      ob[(size_t)o * HW_ + j * 16 + pixl] = (acc[j][r] + sBias[o]) * att;
    }
  }
}

// ---------------------------------------------------------------------------
extern "C" void kernel_launch(void* const* d_in, const int* in_sizes, int n_in,
                              void* d_out, int out_size, void* d_ws, size_t ws_size,
                              hipStream_t stream) {
  const float* x    = (const float*)d_in[0];
  const float* pw_w = (const float*)d_in[1];
  const float* pw_b = (const float*)d_in[2];
  float* out = (float*)d_out;

  float* ws       = (float*)d_ws;
  float* colsum_x = ws;                 // 1024
  float* colsum_y = ws + 1024;          // 1024
  float* w2       = ws + 2048;          // 1024
  float* bias2    = ws + 3072;          // 16
  float* sumsq_x  = ws + 3088;          // 16
  float* sumsq_y  = ws + 3104;          // 16 (contiguous with sumsq_x)
  float* inv_nx   = ws + 3120;          // 16
  float* inv_ny   = ws + 3136;          // 16
  float* score_x  = ws + 4096;          // B*HW
  float* score_y  = score_x + (size_t)B_ * HW_;

  hipMemsetAsync(sumsq_x, 0, 32 * sizeof(float), stream);

  psa_colsum<<<dim3(B_ * C_), 256, 0, stream>>>(x, colsum_x);
  psa_derived<<<1, 1024, 0, stream>>>(pw_w, pw_b, colsum_x, colsum_y, w2, bias2);
  psa_scores<<<dim3(HW_ / 1024, B_), 256, 0, stream>>>(
      x, colsum_x, w2, bias2, score_x, score_y, sumsq_x, sumsq_y);
  psa_norms<<<1, 32, 0, stream>>>(sumsq_x, sumsq_y, inv_nx, inv_ny);
  psa_fused<<<dim3(HW_ / 128, B_), 256, 0, stream>>>(
      x, pw_w, pw_b, score_x, score_y, inv_nx, inv_ny, out);
}